// PLE_25915832664240
// MI455X (gfx1250) — compile-verified
//
#include <hip/hip_runtime.h>
#include <hip/hip_bf16.h>
#include <stdint.h>

typedef __attribute__((ext_vector_type(4))) float v4f;

#define PLE_NBINS 64

__global__ __launch_bounds__(256) void ple_encode_kernel(
    const float* __restrict__ x,
    const float* __restrict__ bins,
    float* __restrict__ out,
    int batch,
    int total_threads)
{
    __shared__ float s_bins[PLE_NBINS + 1];

    const int t = threadIdx.x;

    // ---- Stage the 65 bin edges into LDS via CDNA5 async global->LDS DMA ----
    // One lane per edge; ASYNCcnt tracks completion (not LOADcnt/DScnt).
    if (t <= PLE_NBINS) {
        // addrspace(3) offset: low 32 bits of the generic LDS pointer.
        unsigned lds_addr = (unsigned)(uintptr_t)(&s_bins[t]);
        unsigned long long gaddr = (unsigned long long)(uintptr_t)(bins + t);
        asm volatile("global_load_async_to_lds_b32 %0, %1, off"
                     :
                     : "v"(lds_addr), "v"(gaddr)
                     : "memory");
    }
    asm volatile("s_wait_asynccnt 0x0" ::: "memory");
    __syncthreads();

    // ---- Per-thread loop-invariant bin parameters ----
    // 16 lanes cover one element; lane q owns bins 4q..4q+3.
    const int q = t & 15;
    const float b0 = s_bins[4 * q + 0];
    const float b1 = s_bins[4 * q + 1];
    const float b2 = s_bins[4 * q + 2];
    const float b3 = s_bins[4 * q + 3];
    const float b4 = s_bins[4 * q + 4];

    const float a0 = 1.0f / (b1 - b0);
    const float a1 = 1.0f / (b2 - b1);
    const float a2 = 1.0f / (b3 - b2);
    const float a3 = 1.0f / (b4 - b3);

    // Reference select == clamp(v, L, U) with the clamp released on the
    // grid's outer edges: bin 0 has no lower clamp, bin 63 no upper clamp.
    const float INF = __builtin_inff();
    const float lo0 = (q > 0)  ? 0.0f : -INF;  // bin j=0 only when q==0
    const float hi3 = (q < 15) ? 1.0f :  INF;  // bin j=63 only when q==15

    const int tid     = (int)(blockIdx.x * blockDim.x) + t;
    int       e       = tid >> 4;
    const int estride = total_threads >> 4;

    for (; e < batch; e += estride) {
        // Prefetch next grid-stride element of x (speculative; OOB is dropped).
        __builtin_prefetch(x + e + estride, 0, 0);

        const float xv = x[e];

        v4f r;
        r.x = fminf(fmaxf((xv - b0) * a0, lo0), 1.0f);
        r.y = fminf(fmaxf((xv - b1) * a1, 0.0f), 1.0f);
        r.z = fminf(fmaxf((xv - b2) * a2, 0.0f), 1.0f);
        r.w = fminf(fmaxf((xv - b3) * a3, 0.0f), hi3);

        // 256 MB streaming output >> 192 MB L2: bypass with non-temporal
        // stores (th:TH_STORE_NT). 16B per lane, fully coalesced per wave.
        __builtin_nontemporal_store(
            r, reinterpret_cast<v4f*>(out + (size_t)e * PLE_NBINS + q * 4));
    }
}

extern "C" void kernel_launch(void* const* d_in, const int* in_sizes, int n_in,
                              void* d_out, int out_size, void* d_ws, size_t ws_size,
                              hipStream_t stream) {
    const float* x    = (const float*)d_in[0];
    const float* bins = (const float*)d_in[1];
    float*       out  = (float*)d_out;

    const int batch = in_sizes[0];   // 1,000,000

    const int block = 256;           // 8 wave32 waves per workgroup
    const int grid  = 4096;          // grid-stride persistent threads
    ple_encode_kernel<<<grid, block, 0, stream>>>(x, bins, out, batch,
                                                  grid * block);
}